// Fusion_Spatial_att3_6038724018504
// MI455X (gfx1250) — compile-verified
//
#include <hip/hip_runtime.h>
#include <math.h>

typedef __attribute__((ext_vector_type(2))) float v2f;
typedef __attribute__((ext_vector_type(8))) float v8f;

#define DHW (16 * 128 * 128)   /* 262144 */
#define HW  (128 * 128)        /* 16384 */
#define PWS 36                 /* padded Toeplitz row stride */
#define NQ  98                 /* (c,dz,dy) combinations: 2*7*7 */

__device__ __forceinline__ int refl(int i, int n) {
    if (i < 0) i = -i;
    if (i >= n) i = 2 * n - 2 - i;
    return i;
}

// Kernel 1: collapse the 3 reflect-conv branches + 1x1x1 fusion conv into a
// single combined 2x7x7x5 filter (reflect map is pad-amount independent, so
// the fusion is exact everywhere including borders).
__global__ void combine_weights(const float* __restrict__ w1,
                                const float* __restrict__ w2,
                                const float* __restrict__ w3,
                                const float* __restrict__ wf,
                                float* __restrict__ wcomb) {
    int t = threadIdx.x;
    if (t >= 490) return;
    int dx = t % 5;  int t2 = t / 5;
    int dy = t2 % 7; int t3 = t2 / 7;
    int dz = t3 % 7; int c  = t3 / 7;          // c in {0,1}
    float v = wf[0] * w1[c * 245 + dz * 35 + dy * 5 + dx];
    if (dz >= 1 && dz <= 5 && dy >= 1 && dy <= 5 && dx >= 1 && dx <= 3)
        v += wf[1] * w2[c * 75 + (dz - 1) * 15 + (dy - 1) * 3 + (dx - 1)];
    if (dz >= 2 && dz <= 4 && dy >= 2 && dy <= 4 && dx == 2)
        v += wf[2] * w3[c * 9 + (dz - 2) * 3 + (dy - 2)];
    wcomb[t] = v;
}

// Kernel 2: channel pool -> s_in[B][2][D][H][W]  (mean, max over 32 channels)
__global__ void channel_pool(const float* __restrict__ x, float* __restrict__ s_in) {
    int p = blockIdx.x * blockDim.x + threadIdx.x;   // < 4*DHW
    int b = p >> 18;
    int q = p & (DHW - 1);
    const float* xp = x + (size_t)b * 32 * DHW + q;
    float s = 0.0f, m = -INFINITY;
#pragma unroll
    for (int c = 0; c < 32; ++c) {
        float v = xp[(size_t)c * DHW];
        s += v;
        m = fmaxf(m, v);
    }
    s_in[(size_t)(b * 2 + 0) * DHW + q] = s * (1.0f / 32.0f);
    s_in[(size_t)(b * 2 + 1) * DHW + q] = m;
}

// Kernel 3: combined 2x7x7x5 reflect conv via WMMA f32 16x16x4 (Toeplitz GEMM
// along W), then sigmoid, then scale all 32 channels of x.
// One wave per 16x16 (h,w) output tile at fixed (b,d). 8 waves / block.
// B fragments come from a zero-padded LDS Toeplitz table: a single ds load per
// component, no select chains.
__global__ void __launch_bounds__(256)
spatial_att(const float* __restrict__ x,
            const float* __restrict__ s_in,
            const float* __restrict__ wcomb,
            float* __restrict__ out) {
    __shared__ float pWT[NQ * PWS + 4];   // padded taps: pWT[q*PWS + 15 + d] = W5[d], 0 elsewhere
    __shared__ float plane[8][22 * 20];   // per-wave slab: rows h0-3..h0+18, cols w0-2..w0+17

    for (int t = threadIdx.x; t < NQ * PWS + 4; t += 256) pWT[t] = 0.0f;
    __syncthreads();                       // zero-fill visible before scatter
    for (int t = threadIdx.x; t < 490; t += 256) {
        int q = t / 5, d = t - q * 5;
        pWT[q * PWS + 15 + d] = wcomb[t];
    }
    // (first __syncthreads in the main loop orders the scatter before any use)

    int wv   = threadIdx.x >> 5;
    int lane = threadIdx.x & 31;
    int m15  = lane & 15;       // A-row / D-column index
    int hi   = lane >> 4;       // half-wave selector

    int tile = blockIdx.x * 8 + wv;       // 0..4095
    int tw = tile & 7;  int t1 = tile >> 3;
    int th = t1 & 7;    int t2 = t1 >> 3;
    int d0 = t2 & 15;   int b  = t2 >> 4;
    int h0 = th * 16, w0 = tw * 16;

    v8f acc = {};
    int boff = 15 + 2 * hi - m15;   // per-lane Toeplitz offset: pWT index = q*PWS + boff + kb (+1)

    for (int c = 0; c < 2; ++c) {
        for (int dz = 0; dz < 7; ++dz) {
            int z = refl(d0 + dz - 3, 16);
            const float* sp = s_in + (size_t)(b * 2 + c) * DHW + (size_t)z * HW;
            __syncthreads();   // previous plane fully consumed block-wide
            for (int e = lane; e < 440; e += 32) {
                int i = e / 20, j = e - i * 20;
                int hs = refl(h0 - 3 + i, 128);
                int ws = refl(w0 - 2 + j, 128);
                plane[wv][e] = sp[hs * 128 + ws];
            }
            __syncthreads();
#pragma unroll
            for (int dy = 0; dy < 7; ++dy) {
                int qb = (((c * 7) + dz) * 7 + dy) * PWS + boff;
#pragma unroll
                for (int kb = 0; kb < 20; kb += 4) {
                    // A (16x4 f32): lane m15 holds row m15+dy of plane, k = kb + 2*hi + {0,1}
                    v2f a, bm;
                    int arow = (m15 + dy) * 20 + kb + 2 * hi;
                    a.x = plane[wv][arow];
                    a.y = plane[wv][arow + 1];
                    // B (4x16 f32): banded Toeplitz taps, zero-padded -> plain LDS reads
                    bm.x = pWT[qb + kb];
                    bm.y = pWT[qb + kb + 1];
                    acc = __builtin_amdgcn_wmma_f32_16x16x4_f32(
                        false, a, false, bm, (short)0, acc, false, false);
                }
            }
        }
    }

    // Epilogue: sigmoid + scale x (C/D layout: n = lane&15, m = r + 8*hi)
    int wcol = w0 + m15;
#pragma unroll
    for (int r = 0; r < 8; ++r) {
        int h = h0 + r + 8 * hi;
        float g = 1.0f / (1.0f + __expf(-acc[r]));
        int hw = h * 128 + wcol;
#pragma unroll 4
        for (int cc = 0; cc < 32; ++cc) {
            size_t idx = (size_t)((b * 32 + cc) * 16 + d0) * HW + hw;
            __builtin_nontemporal_store(x[idx] * g, &out[idx]);
        }
    }
}

extern "C" void kernel_launch(void* const* d_in, const int* in_sizes, int n_in,
                              void* d_out, int out_size, void* d_ws, size_t ws_size,
                              hipStream_t stream) {
    const float* x  = (const float*)d_in[0];
    const float* w1 = (const float*)d_in[1];
    const float* w2 = (const float*)d_in[2];
    const float* w3 = (const float*)d_in[3];
    const float* wf = (const float*)d_in[4];
    float* out = (float*)d_out;

    float* s_in  = (float*)d_ws;               // 4*2*DHW floats = 8 MiB
    float* wcomb = s_in + (size_t)8 * DHW;     // 490 floats

    combine_weights<<<1, 512, 0, stream>>>(w1, w2, w3, wf, wcomb);
    channel_pool<<<(4 * DHW) / 256, 256, 0, stream>>>(x, s_in);
    spatial_att<<<512, 256, 0, stream>>>(x, s_in, wcomb, out);
}